// RibonFastLoss_87960930222808
// MI455X (gfx1250) — compile-verified
//
#include <hip/hip_runtime.h>
#include <math.h>

// ---------------------------------------------------------------------------
// Problem constants (match reference)
// ---------------------------------------------------------------------------
#define KB    4
#define KH    384
#define KW    384
#define KHW   (KH * KW)
#define KN    64
#define KSEG  (KN - 1)
#define KKS   17
#define KKR   8
#define KSTEPS 20
#define KPIXBLOCKS (KHW / 256)          // 576 blocks per batch
#define KNBSUM     (KB * KPIXBLOCKS)    // 2304 block partial sums

typedef float v2f __attribute__((ext_vector_type(2)));
typedef float v8f __attribute__((ext_vector_type(8)));

struct FiltW { float g[KKS]; float dg[KKS]; };

// ---------------------------------------------------------------------------
// Kernel 1: build M = I + 0.1*(alpha*D1^T D1 + beta*D2^T D2) and invert it
// with Gauss-Jordan in LDS (matrix is SPD, diagonally dominant: no pivoting).
// One block of 64 threads; thread r owns row r of the [64 x 128] augmented.
// ---------------------------------------------------------------------------
__global__ __launch_bounds__(64) void rfl_ainv_kernel(float* __restrict__ Ainv) {
    __shared__ float S[64][130];
    const int r = threadIdx.x;
    for (int c = 0; c < 64; ++c) {
        float a = 0.f;
        for (int i = 0; i < 63; ++i) {
            float d1r = (r == i) ? -1.f : ((r == i + 1) ? 1.f : 0.f);
            if (d1r != 0.f) {
                float d1c = (c == i) ? -1.f : ((c == i + 1) ? 1.f : 0.f);
                a += 0.01f * d1r * d1c;
            }
        }
        for (int i = 0; i < 62; ++i) {
            float d2r = (r == i) ? 1.f : ((r == i + 1) ? -2.f : ((r == i + 2) ? 1.f : 0.f));
            if (d2r != 0.f) {
                float d2c = (c == i) ? 1.f : ((c == i + 1) ? -2.f : ((c == i + 2) ? 1.f : 0.f));
                a += 0.01f * d2r * d2c;
            }
        }
        S[r][c]      = ((r == c) ? 1.f : 0.f) + 0.1f * a;
        S[r][64 + c] = (r == c) ? 1.f : 0.f;
    }
    __syncthreads();
    for (int k = 0; k < 64; ++k) {
        if (r != k) {
            float f = S[r][k] / S[k][k];
            for (int c = k; c < 128; ++c) S[r][c] -= f * S[k][c];
        }
        __syncthreads();
    }
    float inv = 1.f / S[r][r];
    for (int c = 0; c < 64; ++c) Ainv[r * 64 + c] = S[r][64 + c] * inv;
}

// ---------------------------------------------------------------------------
// Kernels 2-3: separable DoG convolution (zero-padded 'SAME', then *10).
// ch0 = dg_rows (x) conv g_cols (y);  ch1 = g_rows conv dg_cols.
// ---------------------------------------------------------------------------
__global__ __launch_bounds__(256) void rfl_convcol_kernel(const float* __restrict__ pred,
                                                          float* __restrict__ SB,
                                                          float* __restrict__ DB,
                                                          FiltW fw) {
    int idx = blockIdx.x * 256 + threadIdx.x;
    if (idx >= KB * KHW) return;
    int y = idx % KW;
    int x = (idx / KW) % KH;
    int b = idx / KHW;
    const float* p = pred + (size_t)b * KHW + (size_t)x * KW;
    float s = 0.f, d = 0.f;
#pragma unroll
    for (int v = 0; v < KKS; ++v) {
        int yy = y + v - KKR;
        if (yy >= 0 && yy < KW) {
            float val = p[yy];
            s += val * fw.g[v];
            d += val * fw.dg[v];
        }
    }
    SB[idx] = s;
    DB[idx] = d;
}

__global__ __launch_bounds__(256) void rfl_convrow_kernel(const float* __restrict__ SB,
                                                          const float* __restrict__ DB,
                                                          float* __restrict__ GI,
                                                          FiltW fw) {
    int idx = blockIdx.x * 256 + threadIdx.x;
    if (idx >= KB * KHW) return;
    int y = idx % KW;
    int x = (idx / KW) % KH;
    int b = idx / KHW;
    float c0 = 0.f, c1 = 0.f;
#pragma unroll
    for (int u = 0; u < KKS; ++u) {
        int xx = x + u - KKR;
        if (xx >= 0 && xx < KH) {
            size_t o = (size_t)b * KHW + (size_t)xx * KW + y;
            c0 += SB[o] * fw.dg[u];
            c1 += DB[o] * fw.g[u];
        }
    }
    size_t base = (size_t)b * 2 * KHW + (size_t)x * KW + y;
    GI[base]       = 10.0f * c0;   // EXTGRADFAC
    GI[base + KHW] = 10.0f * c1;
}

// ---------------------------------------------------------------------------
// Kernel 4: snake evolution. One wave (32 lanes) per batch sample.
// 20 steps of: bilinear sample force, then pos = Ainv(64x64) @ (pos + 0.1*f)
// where the matvec runs on V_WMMA_F32_16X16X4_F32:
//   4 output M-tiles x 16 K-chunks of 4; N=2 useful columns (rest padded).
// Fragment layouts (wave32, "lane-half selects half of the K/M range"):
//   A: lane L -> M = L&15, VGPR j -> K = 4k + 2*(L>>4) + j
//   B: lane L -> N = L&15, VGPR j -> K = 4k + 2*(L>>4) + j
//   D: VGPR r -> M = 16m + 8*(L>>4) + r, N = L&15
// B columns N>=2 are loaded unconditionally from a zero-padded LDS region:
// matrix-product columns are independent, so those only affect discarded D
// columns -- this keeps EXEC all-ones with no saveexec masking in the loop.
// ---------------------------------------------------------------------------
__global__ __launch_bounds__(32) void rfl_snake_kernel(const float* __restrict__ GI,
                                                       const float* __restrict__ Ainv,
                                                       const float* __restrict__ init,
                                                       float* __restrict__ posOut) {
    __shared__ float sA[64 * 64];
    __shared__ float sp[128];   // pos [node][2]
    __shared__ float sv[160];   // v = pos + 0.1*f, padded (indices up to 141)
    const int lane = threadIdx.x;
    const int b = blockIdx.x;
    for (int i = lane; i < 4096; i += 32) sA[i] = Ainv[i];
    for (int i = lane; i < 128; i += 32) sp[i] = init[b * 128 + i];
    sv[128 + lane] = 0.f;                       // zero the pad region once
    const float* g0 = GI + (size_t)b * 2 * KHW;
    const float* g1 = g0 + KHW;
    const int Ml = lane & 15;
    const int Kh = lane >> 4;
    __syncthreads();

    for (int step = 0; step < KSTEPS; ++step) {
        // ---- bilinear force sampling: lane handles nodes lane and lane+32
#pragma unroll
        for (int h = 0; h < 2; ++h) {
            int n = lane + 32 * h;
            float px = sp[2 * n + 0], py = sp[2 * n + 1];
            float x = fminf(fmaxf(px, 0.f), (float)(KH - 1));
            float y = fminf(fmaxf(py, 0.f), (float)(KW - 1));
            int x0 = (int)floorf(x); x0 = min(max(x0, 0), KH - 2);
            int y0 = (int)floorf(y); y0 = min(max(y0, 0), KW - 2);
            float fx = x - (float)x0, fy = y - (float)y0;
            size_t o = (size_t)x0 * KW + y0;
            float a00 = g0[o], a01 = g0[o + 1], a10 = g0[o + KW], a11 = g0[o + KW + 1];
            float b00 = g1[o], b01 = g1[o + 1], b10 = g1[o + KW], b11 = g1[o + KW + 1];
            float f0 = (a00 * (1.f - fy) + a01 * fy) * (1.f - fx)
                     + (a10 * (1.f - fy) + a11 * fy) * fx;
            float f1 = (b00 * (1.f - fy) + b01 * fy) * (1.f - fx)
                     + (b10 * (1.f - fy) + b11 * fy) * fx;
            sv[2 * n + 0] = px + 0.1f * f0;  // STEPSZ
            sv[2 * n + 1] = py + 0.1f * f1;
        }
        __syncthreads();

        // ---- newpos = Ainv @ v via f32 WMMA (EXEC all-ones throughout)
#pragma unroll
        for (int m = 0; m < 4; ++m) {
            v8f acc = {0.f, 0.f, 0.f, 0.f, 0.f, 0.f, 0.f, 0.f};
#pragma unroll
            for (int k = 0; k < 16; ++k) {
                const int row = 16 * m + Ml;
                const int kk = 4 * k + 2 * Kh;
                v2f a, bb;
                a.x = sA[row * 64 + kk + 0];
                a.y = sA[row * 64 + kk + 1];
                bb.x = sv[(kk + 0) * 2 + Ml];    // unconditional; cols>=2 junk
                bb.y = sv[(kk + 1) * 2 + Ml];
                acc = __builtin_amdgcn_wmma_f32_16x16x4_f32(
                    /*neg_a=*/false, a, /*neg_b=*/false, bb,
                    /*c_mod=*/(short)0, acc, /*reuse_a=*/false, /*reuse_b=*/false);
            }
            if (Ml < 2) {
#pragma unroll
                for (int rr = 0; rr < 8; ++rr)
                    sp[(16 * m + 8 * Kh + rr) * 2 + Ml] = acc[rr];
            }
        }
        __syncthreads();
    }
    for (int i = lane; i < 128; i += 32) posOut[b * 128 + i] = sp[i];
}

// ---------------------------------------------------------------------------
// Kernel 5: radial widths. One wave per (batch, node); lanes split 36 angles.
// hit: in-bounds pixel with sigmoid(-pred)<0.5  <=>  pred>0 (strict).
// At first radius with (hit | oob): width = hit ? r-1 : 0.
// ---------------------------------------------------------------------------
__global__ __launch_bounds__(32) void rfl_widths_kernel(const float* __restrict__ pred,
                                                        const float* __restrict__ pos,
                                                        float* __restrict__ wid) {
    const int gb = blockIdx.x;        // 0 .. KB*64-1
    const int b = gb >> 6;
    const int n = gb & 63;
    const int lane = threadIdx.x;
    const float px = rintf(pos[b * 128 + n * 2 + 0]);
    const float py = rintf(pos[b * 128 + n * 2 + 1]);
    const float* p = pred + (size_t)b * KHW;

    float c0, s0, c1 = 0.f, s1 = 0.f;
    {
        float ang = (float)(lane * 10) * 0.017453292519943295f;
        c0 = cosf(ang); s0 = sinf(ang);
    }
    const bool has2 = (lane + 32) < 36;
    if (has2) {
        float ang = (float)((lane + 32) * 10) * 0.017453292519943295f;
        c1 = cosf(ang); s1 = sinf(ang);
    }

    float width = 0.f;
    for (int r = 1; r < KH; ++r) {
        float rf = (float)r;
        bool hitl = false, oobl = false;
        {
            float xf = floorf(px + rf * c0);
            float yf = floorf(py + rf * s0);
            bool inb = (xf >= 0.f) && (xf < (float)KH) && (yf >= 0.f) && (yf < (float)KW);
            if (!inb) oobl = true;
            int cx = min(max((int)xf, 0), KH - 1);
            int cy = min(max((int)yf, 0), KW - 1);
            if (inb && p[(size_t)cx * KW + cy] > 0.f) hitl = true;
        }
        if (has2) {
            float xf = floorf(px + rf * c1);
            float yf = floorf(py + rf * s1);
            bool inb = (xf >= 0.f) && (xf < (float)KH) && (yf >= 0.f) && (yf < (float)KW);
            if (!inb) oobl = true;
            int cx = min(max((int)xf, 0), KH - 1);
            int cy = min(max((int)yf, 0), KW - 1);
            if (inb && p[(size_t)cx * KW + cy] > 0.f) hitl = true;
        }
        if (__ballot((int)hitl) != 0ull) { width = rf - 1.f; break; }
        if (__ballot((int)oobl) != 0ull) { width = 0.f; break; }
    }
    if (lane == 0) wid[b * 64 + n] = width;
}

// ---------------------------------------------------------------------------
// Kernel 6: per-pixel distance/ribbon maps + fused loss terms; block partial
// sums written to workspace for a deterministic final reduction.
// ---------------------------------------------------------------------------
__global__ __launch_bounds__(256) void rfl_render_kernel(const float* __restrict__ pred,
                                                         const float* __restrict__ pos,
                                                         const float* __restrict__ wid,
                                                         float* __restrict__ bsum) {
    __shared__ float p0x[KSEG], p0y[KSEG], vx[KSEG], vy[KSEG], vv[KSEG], w0[KSEG], dw[KSEG];
    __shared__ float red[256];
    const int b = blockIdx.y;
    const int tid = threadIdx.x;
    if (tid < KSEG) {
        float ax = pos[b * 128 + tid * 2 + 0];
        float ay = pos[b * 128 + tid * 2 + 1];
        float bx = pos[b * 128 + (tid + 1) * 2 + 0];
        float by = pos[b * 128 + (tid + 1) * 2 + 1];
        p0x[tid] = ax; p0y[tid] = ay;
        float vX = bx - ax, vY = by - ay;
        vx[tid] = vX; vy[tid] = vY;
        vv[tid] = vX * vX + vY * vY + 1e-8f;
        float wa = wid[b * 64 + tid], wb = wid[b * 64 + tid + 1];
        w0[tid] = wa; dw[tid] = wb - wa;
    }
    __syncthreads();

    const int pix = blockIdx.x * 256 + tid;
    const float gx = (float)(pix / KW);
    const float gy = (float)(pix % KW);
    float dmin = 3.4e38f, pmin = 3.4e38f;
    for (int s = 0; s < KSEG; ++s) {
        float d0 = gx - p0x[s];
        float d1 = gy - p0y[s];
        float t = (d0 * vx[s] + d1 * vy[s]) / vv[s];
        t = fminf(fmaxf(t, 0.f), 1.f);
        float ex = d0 - t * vx[s];
        float ey = d1 - t * vy[s];
        float dist = sqrtf(ex * ex + ey * ey + 1e-12f);
        dmin = fminf(dmin, dist);
        pmin = fminf(pmin, dist - (w0[s] + t * dw[s]));
    }
    float dmap = fminf(dmin, 15.0f);               // DMAX
    float pmask = (pmin <= 0.f) ? 1.f : 0.f;
    float pr = pred[(size_t)b * KHW + pix];
    float sg = 1.f / (1.f + expf(-pr));
    float e1 = pr - dmap;
    float e2 = sg - pmask;
    red[tid] = e1 * e1 + e2 * e2;
    __syncthreads();
    for (int off = 128; off > 0; off >>= 1) {
        if (tid < off) red[tid] += red[tid + off];
        __syncthreads();
    }
    if (tid == 0) bsum[b * KPIXBLOCKS + blockIdx.x] = red[0];
}

__global__ __launch_bounds__(256) void rfl_final_kernel(const float* __restrict__ bsum,
                                                        float* __restrict__ out) {
    __shared__ float red[256];
    int tid = threadIdx.x;
    float s = 0.f;
    for (int i = tid; i < KNBSUM; i += 256) s += bsum[i];
    red[tid] = s;
    __syncthreads();
    for (int off = 128; off > 0; off >>= 1) {
        if (tid < off) red[tid] += red[tid + off];
        __syncthreads();
    }
    if (tid == 0) out[0] = red[0] * (1.0f / (float)(KB * KHW));
}

// ---------------------------------------------------------------------------
// Launcher
// ---------------------------------------------------------------------------
extern "C" void kernel_launch(void* const* d_in, const int* in_sizes, int n_in,
                              void* d_out, int out_size, void* d_ws, size_t ws_size,
                              hipStream_t stream) {
    (void)in_sizes; (void)n_in; (void)out_size; (void)ws_size;
    const float* pred   = (const float*)d_in[0];  // [B,1,H,W]
    const float* snake0 = (const float*)d_in[1];  // [B,64,2]
    float* out = (float*)d_out;

    float* f    = (float*)d_ws;
    float* AINV = f;                         size_t off = 4096;
    float* SB   = f + off;                   off += (size_t)KB * KHW;
    float* DB   = f + off;                   off += (size_t)KB * KHW;
    float* GI   = f + off;                   off += (size_t)KB * 2 * KHW;
    float* POS  = f + off;                   off += KB * 128;
    float* WID  = f + off;                   off += KB * 64;
    float* BSUM = f + off;                   off += KNBSUM;

    // Gaussian + derivative-of-Gaussian taps (std=2, r=8), computed on host,
    // passed by value (tiny, graph-capture safe).
    FiltW fw;
    {
        float tmp[KKS]; float sum = 0.f;
        for (int i = 0; i < KKS; ++i) {
            float x = (float)(i - KKR);
            float e = expf(-0.5f * (x * 0.5f) * (x * 0.5f));
            tmp[i] = e; sum += e;
            fw.dg[i] = -(x * 0.25f) * e;
        }
        for (int i = 0; i < KKS; ++i) fw.g[i] = tmp[i] / sum;
    }

    const int nPix = KB * KHW;
    const int blocks = (nPix + 255) / 256;
    hipLaunchKernelGGL(rfl_ainv_kernel,    dim3(1),               dim3(64),  0, stream, AINV);
    hipLaunchKernelGGL(rfl_convcol_kernel, dim3(blocks),          dim3(256), 0, stream, pred, SB, DB, fw);
    hipLaunchKernelGGL(rfl_convrow_kernel, dim3(blocks),          dim3(256), 0, stream, SB, DB, GI, fw);
    hipLaunchKernelGGL(rfl_snake_kernel,   dim3(KB),              dim3(32),  0, stream, GI, AINV, snake0, POS);
    hipLaunchKernelGGL(rfl_widths_kernel,  dim3(KB * 64),         dim3(32),  0, stream, pred, POS, WID);
    hipLaunchKernelGGL(rfl_render_kernel,  dim3(KPIXBLOCKS, KB),  dim3(256), 0, stream, pred, POS, WID, BSUM);
    hipLaunchKernelGGL(rfl_final_kernel,   dim3(1),               dim3(256), 0, stream, BSUM, out);
}